// SineGen_78932908966636
// MI455X (gfx1250) — compile-verified
//
#include <hip/hip_runtime.h>
#include <math.h>

#define SAMP_RATE_  24000
#define UPSAMPLE_   300
#define DIM_        9
#define NB_         32
#define NL_         120000
#define NF_         400                 // NL_/UPSAMPLE_ frames
#define SINE_AMP_   0.1f
#define NOISE_STD_  0.003f
#define TWO_PI_     6.28318530717958647692f
#define TILE_       256

// ---------- CDNA5 async global->LDS path (gfx1250) -------------------------
#if defined(__gfx1250__) && __has_builtin(__builtin_amdgcn_global_load_async_to_lds_b128)
#define HAVE_ASYNC_LDS 1
#else
#define HAVE_ASYNC_LDS 0
#endif

typedef int   v4i __attribute__((ext_vector_type(4)));
typedef float v4f __attribute__((ext_vector_type(4)));
typedef __attribute__((address_space(1))) v4i* gptr_v4i;
typedef __attribute__((address_space(3))) v4i* lptr_v4i;

__device__ __forceinline__ void async_copy_b128(const float* gsrc, float* ldst) {
#if HAVE_ASYNC_LDS
    // 16B per lane, memory -> LDS, tracked by ASYNCcnt (no VGPR data path)
    __builtin_amdgcn_global_load_async_to_lds_b128(
        (gptr_v4i)(gsrc),
        (lptr_v4i)(ldst),
        0 /*imm offset*/, 0 /*cpol*/);
#else
    *(v4f*)ldst = *(const v4f*)gsrc;
#endif
}

__device__ __forceinline__ void wait_async_lds() {
#if HAVE_ASYNC_LDS
#if __has_builtin(__builtin_amdgcn_s_wait_asynccnt)
    __builtin_amdgcn_s_wait_asynccnt(0);
#else
    asm volatile("s_wait_asynccnt 0" ::: "memory");
#endif
#endif
}

// ---------- Pass 1: per-(b,harmonic) phase cumsum over 400 frames ----------
// 288 independent serial scans; writes phase[(b*9+d)*400 + f] = cumsum*2pi.
__global__ void SineGen_phase_scan(const float* __restrict__ f0,
                                   float* __restrict__ phase) {
    int idx = blockIdx.x * blockDim.x + threadIdx.x;
    if (idx >= NB_ * DIM_) return;
    const int b = idx / DIM_;
    const int d = idx % DIM_;
    const float h = (float)(d + 1);
    const float* fp = f0 + (long long)b * NL_;
    float* pp = phase + (long long)idx * NF_;
    float s = 0.0f;
    for (int f = 0; f < NF_; ++f) {
        float r = fmodf(fp[f * UPSAMPLE_] * h * (1.0f / SAMP_RATE_), 1.0f);
        s += r;
        pp[f] = s * TWO_PI_;
    }
}

// ---------- Pass 2: main streaming kernel ----------------------------------
// One block = 256 consecutive samples of one batch row.
__global__ __launch_bounds__(TILE_) void SineGen_main(
        const float* __restrict__ f0,
        const float* __restrict__ noise,
        const float* __restrict__ phase,
        float* __restrict__ out_sine,
        float* __restrict__ out_uv,
        float* __restrict__ out_ns) {
    __shared__ __align__(16) float ldsN[TILE_ * DIM_];   // noise in, noise_scaled out
    __shared__ __align__(16) float ldsO[TILE_ * DIM_];   // sine_waves out

    const int tid    = threadIdx.x;
    const int b      = blockIdx.y;
    const int l_base = blockIdx.x * TILE_;
    const int rem    = NL_ - l_base;
    const int count  = (rem < TILE_) ? rem : TILE_;      // 256 or 192; both %4==0
    const int nvec   = (count * DIM_) >> 2;              // 128-bit vectors in tile

    const long long base9 = ((long long)b * NL_ + l_base) * DIM_;  // float index

    // Stage the stride-9 noise tile into LDS with coalesced async b128 DMA.
    const float* gN = noise + base9;
    for (int v = tid; v < nvec; v += TILE_)
        async_copy_b128(gN + 4 * v, ldsN + 4 * v);
    wait_async_lds();
    __syncthreads();

    if (tid < count) {
        const int l = l_base + tid;
        const float f0v = __builtin_nontemporal_load(f0 + (long long)b * NL_ + l);
        const float uv  = (f0v > 0.0f) ? 1.0f : 0.0f;
        __builtin_nontemporal_store(uv, out_uv + (long long)b * NL_ + l);

        const float namp = uv * NOISE_STD_ + (1.0f - uv) * (SINE_AMP_ / 3.0f);
        const float suv  = SINE_AMP_ * uv;

        const int  k    = l / UPSAMPLE_;
        const float j   = (float)(l - k * UPSAMPLE_);
        const bool tail = (k >= NF_ - 1);                // last 300 samples: flat phase
        const float* ph = phase + (long long)b * DIM_ * NF_;

        #pragma unroll
        for (int d = 0; d < DIM_; ++d) {
            const float p0 = ph[d * NF_ + (tail ? (NF_ - 1) : k)];
            float pv;
            if (tail) {
                pv = p0;
            } else {
                const float p1 = ph[d * NF_ + k + 1];
                pv = p0 + (p1 - p0) * (1.0f / UPSAMPLE_) * j;
            }
            const float sine = __sinf(pv);
            const float ns   = namp * ldsN[tid * DIM_ + d];  // stride 9: conflict-free
            ldsN[tid * DIM_ + d] = ns;                        // noise_scaled
            ldsO[tid * DIM_ + d] = sine * suv + ns;           // sine_waves
        }
    }
    __syncthreads();

    // Coalesced 128-bit non-temporal flush of both stride-9 outputs
    // (write-once streams: TH=NT keeps L2 for the reused phase table / f0).
    const v4f* vN = (const v4f*)ldsN;
    const v4f* vO = (const v4f*)ldsO;
    v4f* gS  = (v4f*)(out_sine + base9);
    v4f* gNS = (v4f*)(out_ns + base9);
    for (int v = tid; v < nvec; v += TILE_) {
        __builtin_nontemporal_store(vO[v], gS + v);
        __builtin_nontemporal_store(vN[v], gNS + v);
    }
}

// ---------- Host launcher ---------------------------------------------------
extern "C" void kernel_launch(void* const* d_in, const int* in_sizes, int n_in,
                              void* d_out, int out_size, void* d_ws, size_t ws_size,
                              hipStream_t stream) {
    (void)in_sizes; (void)n_in; (void)out_size; (void)ws_size;
    const float* f0    = (const float*)d_in[0];   // (32,120000,1)
    const float* noise = (const float*)d_in[1];   // (32,120000,9)

    float* phase = (float*)d_ws;                  // 32*9*400 floats = 460800 B

    float* out      = (float*)d_out;
    float* out_sine = out;                                        // (32,120000,9)
    float* out_uv   = out + (long long)NB_ * NL_ * DIM_;          // (32,120000)
    float* out_ns   = out_uv + (long long)NB_ * NL_;              // (32,120000,9)

    SineGen_phase_scan<<<dim3((NB_ * DIM_ + 63) / 64), dim3(64), 0, stream>>>(f0, phase);

    dim3 grid((NL_ + TILE_ - 1) / TILE_, NB_);    // 469 x 32 blocks
    SineGen_main<<<grid, dim3(TILE_), 0, stream>>>(f0, noise, phase,
                                                   out_sine, out_uv, out_ns);
}